// AttnPainter_27041114095713
// MI455X (gfx1250) — compile-verified
//
#include <hip/hip_runtime.h>
#include <hip/hip_bf16.h>
#include <stdint.h>

// Problem constants from the reference: B=8, N=256, W=128, K=10
#define BB      8
#define NN      256
#define WW      128
#define KK      10
#define NPIX    (WW * WW)        // 16384 pixels per image
#define TPB     256              // 8 wave32 per block
#define PIX     256              // pixels per block tile (1 per thread)
#define NC      16               // n-slices per TDM chunk (>= K so typical inputs finish in chunk 0)
#define NCHUNKS (NN / NC)        // 16

#if defined(__has_builtin)
#if __has_builtin(__builtin_amdgcn_tensor_load_to_lds) && __has_builtin(__builtin_amdgcn_s_wait_tensorcnt)
#define HAVE_TDM 1
#endif
#endif
#ifndef HAVE_TDM
#define HAVE_TDM 0
#endif

typedef __attribute__((ext_vector_type(4))) unsigned int u32x4;
typedef __attribute__((ext_vector_type(8))) int          i32x8;
typedef __attribute__((ext_vector_type(4))) int          i32x4;

#if HAVE_TDM
// Issue a 2-D TDM load: tile_d1 rows of tile_d0 fp32 elements, row stride
// d0_stride_elems, from gptr into LDS at lds_byte_off.
// D# packing per CDNA5 ISA ch.8 (groups 0..3), data_size=4B, no padding,
// no iteration, no gather, no multicast (not in a cluster).
// amdgpu-toolchain (clang-23) builtin arity is 6: the 5th arg (int32x8) is
// not a documented D# group; zero-filled per the probe-verified pattern.
__device__ __forceinline__ void tdm_load_2d_f32(
    uint32_t lds_byte_off, const float* gptr,
    uint32_t tensor_d0, uint32_t tensor_d1,
    uint32_t tile_d0,   uint32_t tile_d1,
    uint64_t d0_stride_elems)
{
    const uint64_t ga = (uint64_t)(uintptr_t)gptr;

    u32x4 g0;
    g0[0] = 0x1u;                                            // count=1 (valid), user mode
    g0[1] = lds_byte_off;                                    // lds_addr [63:32]
    g0[2] = (uint32_t)(ga & 0xFFFFFFFFull);                  // global_addr [95:64]
    g0[3] = (uint32_t)((ga >> 32) & 0x01FFFFFFull)           // global_addr [120:96]
          | (2u << 30);                                      // type=2 ("image") [127:126]

    i32x8 g1;
    g1[0] = (int)(2u << 16);                                 // data_size=2 (4 bytes); mask/flags=0
    g1[1] = (int)((tensor_d0 & 0xFFFFu) << 16);              // tensor_dim0[15:0] @ bits 63:48
    g1[2] = (int)(((tensor_d0 >> 16) & 0xFFFFu)              // tensor_dim0[31:16]
          |      ((tensor_d1 & 0xFFFFu) << 16));             // tensor_dim1[15:0]
    g1[3] = (int)(((tensor_d1 >> 16) & 0xFFFFu)              // tensor_dim1[31:16]
          |      ((tile_d0 & 0xFFFFu) << 16));               // tile_dim0
    g1[4] = (int)(tile_d1 & 0xFFFFu);                        // tile_dim1 (tile_dim2=0 unused)
    g1[5] = (int)(uint32_t)(d0_stride_elems & 0xFFFFFFFFull);// tensor_dim0_stride[31:0]
    g1[6] = (int)(uint32_t)((d0_stride_elems >> 32) & 0xFFFFull); // stride[47:32]; dim1_stride lo=0
    g1[7] = 0;                                               // dim1_stride hi (unused for 2-D tile)

    i32x4 g2 = {0, 0, 0, 0};                                 // tensor_dim2/3 etc. unused
    i32x4 g3 = {0, 0, 0, 0};
    i32x8 g4 = {0, 0, 0, 0, 0, 0, 0, 0};                     // extra arg on clang-23 lane

    __builtin_amdgcn_tensor_load_to_lds(g0, g1, g2, g3, g4, /*cpol*/0);
}
#endif

__global__ __launch_bounds__(TPB)
void AttnPainter_27041114095713_kernel(const float* __restrict__ alpha,
                                       const float* __restrict__ colors,
                                       float* __restrict__ out)
{
    __shared__ float sbuf[2][NC * PIX];   // double-buffered alpha chunks (2 x 16 KB)
    __shared__ float scol[3 * NN];        // color table for this image (3 KB)

    const int tid  = threadIdx.x;
    const int tilesPerImg = NPIX / PIX;                 // 64
    const int b    = blockIdx.x / tilesPerImg;
    const int pix0 = (blockIdx.x % tilesPerImg) * PIX;

    // Stage colors[b, :, :] -> LDS (layout matches flat (n,ch) order).
    for (int i = tid; i < 3 * NN; i += TPB)
        scol[i] = colors[b * 3 * NN + i];

    const float* abase = alpha + ((size_t)b * NN) * NPIX + pix0;

#if HAVE_TDM
    // Prefetch chunk 0 (the top NC slices of n) into sbuf[0]. One wave issues
    // the DMA for the whole block (TDM ignores EXEC; per-wave issue would
    // duplicate traffic 8x).
    if (tid < 32) {
        tdm_load_2d_f32((uint32_t)(uintptr_t)&sbuf[0][0],
                        abase + (size_t)(NN - NC) * NPIX,
                        NPIX, NN, PIX, NC, NPIX);
    }
#endif

    // Front-to-back compositing state: final canvas[c] = T + A[c].
    float T = 1.0f, A0 = 0.0f, A1 = 0.0f, A2 = 0.0f;
    int cnt = 0;

    for (int c = 0; c < NCHUNKS; ++c) {
        const int par = c & 1;

#if HAVE_TDM
        if (c + 1 < NCHUNKS) {
            // The buffer we prefetch into was last read in iteration c-1;
            // the __syncthreads_and at the end of that iteration fenced it.
            if (tid < 32) {
                tdm_load_2d_f32((uint32_t)(uintptr_t)&sbuf[par ^ 1][0],
                                abase + (size_t)(NN - (size_t)(c + 2) * NC) * NPIX,
                                NPIX, NN, PIX, NC, NPIX);
            }
            // TDM completes in order within a wave: <=1 outstanding means
            // chunk c has landed in LDS. No-op for waves that issued nothing.
            __builtin_amdgcn_s_wait_tensorcnt(1);
        } else {
            __builtin_amdgcn_s_wait_tensorcnt(0);
        }
        __syncthreads();   // publish wave0's completed chunk (and scol on c==0)
#else
        // Fallback: cooperative global->LDS copy of chunk c.
        {
            const int nlo_f = NN - (c + 1) * NC;
            for (int i = tid; i < NC * PIX; i += TPB)
                sbuf[par][i] = abase[(size_t)(nlo_f + i / PIX) * NPIX + (i % PIX)];
        }
        __syncthreads();
#endif

        const int nlo = NN - (c + 1) * NC;
        #pragma unroll
        for (int r = NC - 1; r >= 0; --r) {           // descending n within chunk
            const float a = sbuf[par][r * PIX + tid]; // conflict-free: consecutive lanes
            const float p = 1.0f - a;                 // pred, exactly as reference
            if (p > 0.0f && cnt < KK) {               // top-K mask: K largest n with pred>0
                const int n = nlo + r;                // uniform per iteration -> LDS broadcast
                const float w = T * p;
                A0 += w * scol[n * 3 + 0];
                A1 += w * scol[n * 3 + 1];
                A2 += w * scol[n * 3 + 2];
                T *= (1.0f - p);
                ++cnt;
            }
        }

        // Block-uniform early exit: with uniform-[0,1) alpha every pixel hits
        // K strokes inside chunk 0, cutting traffic from 128 MiB to ~16 MiB.
        if (__syncthreads_and(cnt >= KK)) break;
    }

    const size_t pe = (size_t)pix0 + (size_t)tid;
    out[((size_t)b * 3 + 0) * NPIX + pe] = T + A0;    // white background: T*1 + Acc
    out[((size_t)b * 3 + 1) * NPIX + pe] = T + A1;
    out[((size_t)b * 3 + 2) * NPIX + pe] = T + A2;
}

extern "C" void kernel_launch(void* const* d_in, const int* in_sizes, int n_in,
                              void* d_out, int out_size, void* d_ws, size_t ws_size,
                              hipStream_t stream) {
    const float* alpha  = (const float*)d_in[0];   // (B, N, W, W) fp32
    const float* colors = (const float*)d_in[1];   // (B, N, 3)   fp32
    float* out = (float*)d_out;                    // (B, 3, W, W) fp32

    const dim3 grid(BB * (NPIX / PIX));            // 512 blocks
    const dim3 block(TPB);                         // 256 threads = 8 wave32
    AttnPainter_27041114095713_kernel<<<grid, block, 0, stream>>>(alpha, colors, out);
}